// Embedder_46497315946942
// MI455X (gfx1250) — compile-verified
//
#include <hip/hip_runtime.h>

// 4-D multiresolution hash-grid encoder (Instant-NGP style), CDNA5/gfx1250.
//
// Roofline: ~268 MB of data-dependent 8-byte gathers from ~32 MB of tables
// (fully L2-resident; 192 MB L2) + ~27 MB HBM streaming; ~0.6 GFLOP total.
// => gather/VALU bound; WMMA/TDM structurally inapplicable (per-lane random
// rows, no shared operand matrix). CDNA5 levers used instead:
//   - wave32-uniform level (blockIdx.y) -> scalar dense/hash branch, SGPR
//     level constants, zero divergence
//   - global_load_b64 scale_offset gathers (no per-lane 64-bit address math),
//     all 16 corner loads in flight before accumulation
//   - pass-1 hoists the per-point normalization divisions (exactly the
//     reference's once-per-point computation); its input stream is loaded
//     non-temporally (native clang vector type) to keep table lines resident
//   - factored XOR / index / weight trees (bit-identical associations)

#define NPTS      131072
#define NLEVELS   16
#define FPER      2
#define TBL       262147
#define STARTHASH 2
#define LVL1_OFF  65536       // 16^4 rows before level-1 dense block
#define TWO32_MOD_TBL 212995u // 2^32 mod 262147

typedef float f32x4 __attribute__((ext_vector_type(4)));  // native vector for NT load

// ENTRIES_NUM[i] = int(16 * 1.38**i)
__constant__ int c_num[NLEVELS] = {
    16, 22, 30, 42, 58, 80, 110, 152,
    210, 290, 400, 553, 763, 1053, 1453, 2005
};
// ENTRIES_SIZE[i] = float32(1.0/(num-1)); compile-time double->float rounding
// matches numpy exactly.
__constant__ float c_size[NLEVELS] = {
    (float)(1.0/15.0),   (float)(1.0/21.0),   (float)(1.0/29.0),   (float)(1.0/41.0),
    (float)(1.0/57.0),   (float)(1.0/79.0),   (float)(1.0/109.0),  (float)(1.0/151.0),
    (float)(1.0/209.0),  (float)(1.0/289.0),  (float)(1.0/399.0),  (float)(1.0/552.0),
    (float)(1.0/762.0),  (float)(1.0/1052.0), (float)(1.0/1452.0), (float)(1.0/2004.0)
};

// ---------------- Pass 1: per-point normalization ----------------
__global__ __launch_bounds__(256)
void normalize_pts(const float* __restrict__ xyz,
                   const float* __restrict__ bounds,
                   float* __restrict__ xn_out)
{
    const int n = blockIdx.x * blockDim.x + threadIdx.x;
    // Read-once stream: non-temporal load (th:TH_LOAD_NT), don't displace tables.
    const f32x4 p = __builtin_nontemporal_load(
        reinterpret_cast<const f32x4*>(xyz + (size_t)n * 4));
    f32x4 r;
    r.x = (p.x - bounds[0]) / (bounds[4] - bounds[0]);
    r.y = (p.y - bounds[1]) / (bounds[5] - bounds[1]);
    r.z = (p.z - bounds[2]) / (bounds[6] - bounds[2]);
    r.w = (p.w - bounds[3]) / (bounds[7] - bounds[3]);
    *reinterpret_cast<f32x4*>(xn_out + (size_t)n * 4) = r;   // RT: re-read by pass 2
}

// ---------------- Pass 2: encode ----------------
template <bool PRENORM>
__global__ __launch_bounds__(256)
void embedder_fwd(const float* __restrict__ pts,   // xn if PRENORM, else xyz
                  const float* __restrict__ dense,
                  const float* __restrict__ hash_table,
                  const float* __restrict__ bounds,
                  float* __restrict__ out)
{
    const int l = blockIdx.y;                       // wave-uniform level
    const int n = blockIdx.x * blockDim.x + threadIdx.x;

    const float4 p = *reinterpret_cast<const float4*>(pts + (size_t)n * 4);
    const float pf[4] = {p.x, p.y, p.z, p.w};

    const int   num = c_num[l];
    const float sz  = c_size[l];

    // Per-dimension low/high corner coords and linear weights.
    int   c0[4], c1[4];
    float w0[4], w1[4];
#pragma unroll
    for (int d = 0; d < 4; ++d) {
        float xn;
        if (PRENORM) {
            xn = pf[d];
        } else {
            const float b0 = bounds[d], b1 = bounds[4 + d];
            xn = (pf[d] - b0) / (b1 - b0);            // reference: true division
        }
        const float flt = xn / sz;                    // reference: x / size (f32)
        int lo = (int)flt;                            // trunc; flt >= 0
        int hi = (int)(flt + 1.0f);                   // NOT lo+1: f32 add may round
        const int e = num - 1;
        lo = lo < e ? lo : e;
        hi = hi < e ? hi : e;
        c0[d] = lo;
        c1[d] = hi;
        const float off = flt - (float)lo;            // off relative to clipped c0
        w1[d] = off;
        w0[d] = 1.0f - off;
    }

    // Gather all 16 corner rows (float2 -> global_load_b64 scale_offset).
    float2 v[16];
    if (l < STARTHASH) {
        const float2* tb = reinterpret_cast<const float2*>(dense)
                         + (l == 0 ? 0 : LVL1_OFF);
        const int n2 = num * num, n3 = n2 * num;
        // Factored index tree: pairwise partial sums.
        int t01[4], t23[4];
#pragma unroll
        for (int j = 0; j < 4; ++j) {
            t01[j] = ((j & 1) ? c1[0] : c0[0]) * n3 + ((j & 2) ? c1[1] : c0[1]) * n2;
            t23[j] = ((j & 1) ? c1[2] : c0[2]) * num + ((j & 2) ? c1[3] : c0[3]);
        }
#pragma unroll
        for (int c = 0; c < 16; ++c) {
            const unsigned ind = (unsigned)(t01[c & 3] + t23[c >> 2]);
            v[c] = tb[ind];                            // scale_offset b64 gather
        }
    } else {
        const float2* tb = reinterpret_cast<const float2*>(hash_table)
                         + (size_t)(l - STARTHASH) * TBL;
        // Factored XOR tree over the 8 hoisted 64-bit products (< 2^38 each).
        unsigned long long p01[4], p23[4];
#pragma unroll
        for (int j = 0; j < 4; ++j) {
            p01[j] = ((unsigned long long)((j & 1) ? c1[0] : c0[0]) * 1ull)
                   ^ ((unsigned long long)((j & 2) ? c1[1] : c0[1]) * 19349663ull);
            p23[j] = ((unsigned long long)((j & 1) ? c1[2] : c0[2]) * 51471207ull)
                   ^ ((unsigned long long)((j & 2) ? c1[3] : c0[3]) * 83492791ull);
        }
#pragma unroll
        for (int c = 0; c < 16; ++c) {
            const unsigned long long h = p01[c & 3] ^ p23[c >> 2];
            // h < 2^38: h mod TBL via 2^32 ≡ 212995 (mod 262147).
            const unsigned int hi32 = (unsigned int)(h >> 32);   // < 64
            const unsigned int lo32 = (unsigned int)h;
            unsigned int r = hi32 * TWO32_MOD_TBL + (lo32 % (unsigned int)TBL);
            r %= (unsigned int)TBL;                              // fits 32b, no overflow
            v[c] = tb[r];                              // scale_offset b64 gather
        }
    }

    // Factored weight tree; association matches reference (m0*m1)*m2.
    float w01[4], tri[8];
#pragma unroll
    for (int j = 0; j < 4; ++j)
        w01[j] = ((j & 1) ? w1[0] : w0[0]) * ((j & 2) ? w1[1] : w0[1]);
#pragma unroll
    for (int k = 0; k < 8; ++k)
        tri[k] = w01[k & 3] * ((k & 4) ? w1[2] : w0[2]);

    // tri is identical for corners c and c+8 (they differ only in dim 3).
    float s_ta = 0.0f, s_tb = 0.0f;
#pragma unroll
    for (int c = 0; c < 8; ++c)
        s_ta += tri[c] * (v[c].x + v[c].y);
#pragma unroll
    for (int c = 8; c < 16; ++c)
        s_tb += tri[c - 8] * (v[c].x + v[c].y);
    const float s_four = w0[3] * s_ta + w1[3] * s_tb;  // exact blend identity

    // interp: (N,16) row-major; smooth_t: (N,32) = [ta(16) | tb(16)].
    out[(size_t)n * NLEVELS + l] = s_four;
    float* smooth = out + (size_t)NPTS * NLEVELS;
    smooth[(size_t)n * 32 + l]      = s_ta;
    smooth[(size_t)n * 32 + 16 + l] = s_tb;
}

extern "C" void kernel_launch(void* const* d_in, const int* in_sizes, int n_in,
                              void* d_out, int out_size, void* d_ws, size_t ws_size,
                              hipStream_t stream) {
    (void)in_sizes; (void)n_in; (void)out_size;
    const float* xyz        = (const float*)d_in[0];   // (131072, 4) f32
    const float* dense      = (const float*)d_in[1];   // (299792, 2) f32
    const float* hash_table = (const float*)d_in[2];   // (14, 262147, 2) f32
    const float* bounds     = (const float*)d_in[3];   // (2, 4) f32
    float* out = (float*)d_out;                        // 131072*(16+32) f32

    dim3 block(256);
    dim3 grid(NPTS / 256, NLEVELS);                    // 512 x 16 blocks

    const size_t xn_bytes = (size_t)NPTS * 4 * sizeof(float);
    if (ws_size >= xn_bytes) {
        // Two-pass: hoist the 4 normalization divisions out of the 16-level loop.
        float* xn = (float*)d_ws;
        normalize_pts<<<dim3(NPTS / 256), block, 0, stream>>>(xyz, bounds, xn);
        embedder_fwd<true><<<grid, block, 0, stream>>>(xn, dense, hash_table, bounds, out);
    } else {
        embedder_fwd<false><<<grid, block, 0, stream>>>(xyz, dense, hash_table, bounds, out);
    }
}